// R_GNN_Model_78417512891194
// MI455X (gfx1250) — compile-verified
//
#include <hip/hip_runtime.h>

typedef float v2f __attribute__((ext_vector_type(2)));
typedef float v8f __attribute__((ext_vector_type(8)));

#define NN_NODES 100000
#define NN_EDGES 1600000
#define DIM_IN 128
#define DIM_HID 128
#define DIM_OUT 64
#define N_REL 40
#define N_BASES 31

// ---------------------------------------------------------------------------
// W[r] = sum_b comp[r,b] * basis[b]   (flattened over in*out)
// ---------------------------------------------------------------------------
__global__ void combine_weights(const float* __restrict__ comp,
                                const float* __restrict__ basis,
                                float* __restrict__ W,
                                int nrel, int nb, int inout) {
  int idx = blockIdx.x * blockDim.x + threadIdx.x;
  if (idx >= nrel * inout) return;
  int r = idx / inout;
  int io = idx - r * inout;
  float acc = 0.f;
  for (int b = 0; b < nb; ++b)
    acc += comp[r * nb + b] * basis[(size_t)b * inout + io];
  W[idx] = acc;
}

// ---------------------------------------------------------------------------
// counts / compaction
// ---------------------------------------------------------------------------
__global__ void zero_i32(int* p, int n) {
  int i = blockIdx.x * blockDim.x + threadIdx.x;
  if (i < n) p[i] = 0;
}

__global__ void count_edges(const int* __restrict__ dst,
                            const int* __restrict__ etype,
                            int* cnt, int E, int nnodes) {
  int e = blockIdx.x * blockDim.x + threadIdx.x;
  if (e >= E) return;
  atomicAdd(&cnt[(size_t)etype[e] * nnodes + dst[e]], 1);
}

// For each (relation, node): if count>0, assign a compact row index, record
// node in `list`, reciprocal in `invc` (in place of the int counts).
__global__ void build_compact(float* invc /* aliases int counts */,
                              int* __restrict__ map,
                              int* __restrict__ list,
                              int* __restrict__ nnz,
                              int nnodes, int total) {
  int i = blockIdx.x * blockDim.x + threadIdx.x;
  if (i >= total) return;
  int c = ((int*)invc)[i];
  int r = i / nnodes;
  int n = i - r * nnodes;
  if (c > 0) {
    int idx = atomicAdd(&nnz[r], 1);
    map[i] = idx;
    list[(size_t)r * nnodes + idx] = n;
    invc[i] = 1.0f / (float)c;
  } else {
    map[i] = 0;
    invc[i] = 0.0f;
  }
}

// zero compacted agg rows up to next multiple of 128 rows (for unguarded A loads)
__global__ void zero_agg(float* agg, const int* __restrict__ nnzPtr) {
  int i = blockIdx.x * blockDim.x + threadIdx.x;
  int nr = (*nnzPtr + 127) & ~127;
  if (i < nr * 128) agg[i] = 0.f;
}

__global__ void relu_k(float* p, int n) {
  int i = blockIdx.x * blockDim.x + threadIdx.x;
  if (i < n) p[i] = fmaxf(p[i], 0.f);
}

// ---------------------------------------------------------------------------
// scatter: agg[map[dst]] += x[src] for edges of relation `rel`.
// 32 lanes per edge, float4 per lane (128 floats / row).
// ---------------------------------------------------------------------------
__global__ __launch_bounds__(1024) void scatter_rel(
    const int* __restrict__ src, const int* __restrict__ dst,
    const int* __restrict__ etype, const int* __restrict__ map,
    const float* __restrict__ x, float* agg, int E, int rel) {
  int e = blockIdx.x * 32 + (threadIdx.x >> 5);
  if (e >= E) return;
  if (etype[e] != rel) return;
  int lane = threadIdx.x & 31;
  const float4 v = *(const float4*)(x + (size_t)src[e] * 128 + lane * 4);
  int crow = map[(size_t)rel * NN_NODES + dst[e]];
  float* a = agg + (size_t)crow * 128 + lane * 4;
  atomicAdd(a + 0, v.x);
  atomicAdd(a + 1, v.y);
  atomicAdd(a + 2, v.z);
  atomicAdd(a + 3, v.w);
}

// ---------------------------------------------------------------------------
// Shared WMMA tile core. B is staged in LDS k-pair-interleaved:
//   Bs[k>>1][n] = (B[k][n], B[k+1][n])  -> each frag is ONE ds_load_b64.
// Row stride 80 float2s => half-wave groups hit disjoint 32-bank sets.
// ---------------------------------------------------------------------------
#define BS_STRIDE 80

__device__ __forceinline__ void load_b_slab(float2 (*Bs)[BS_STRIDE],
                                            const float* __restrict__ B,
                                            int colBase, int ldb, int tid) {
  for (int i = tid; i < 128 * 16; i += 256) {
    int r = i >> 4;
    int c4 = (i & 15) << 2;
    const float4 v = *(const float4*)(B + (size_t)r * ldb + colBase + c4);
    float* d0 = ((float*)&Bs[r >> 1][c4]) + (r & 1);
    d0[0] = v.x;
    d0[2] = v.y;
    d0[4] = v.z;
    d0[6] = v.w;
  }
}

__device__ __forceinline__ void wmma_tile(const float* __restrict__ Arow,
                                          const float2 (*Bs)[BS_STRIDE],
                                          int laneLo, int laneHi,
                                          v8f& acc0, v8f& acc1, v8f& acc2, v8f& acc3) {
#pragma unroll 4
  for (int kk = 0; kk < 128; kk += 4) {
    const int p = (kk >> 1) + laneHi;
    const v2f a = *(const v2f*)(Arow + kk + (laneHi << 1));
    const v2f b0 = *(const v2f*)(&Bs[p][laneLo + 0]);
    const v2f b1 = *(const v2f*)(&Bs[p][laneLo + 16]);
    const v2f b2 = *(const v2f*)(&Bs[p][laneLo + 32]);
    const v2f b3 = *(const v2f*)(&Bs[p][laneLo + 48]);
    acc0 = __builtin_amdgcn_wmma_f32_16x16x4_f32(false, a, false, b0, (short)0, acc0, false, false);
    acc1 = __builtin_amdgcn_wmma_f32_16x16x4_f32(false, a, false, b1, (short)0, acc1, false, false);
    acc2 = __builtin_amdgcn_wmma_f32_16x16x4_f32(false, a, false, b2, (short)0, acc2, false, false);
    acc3 = __builtin_amdgcn_wmma_f32_16x16x4_f32(false, a, false, b3, (short)0, acc3, false, false);
  }
}

// ---------------------------------------------------------------------------
// Dense root GEMM:  C[M x ldb] = A[M x 128] @ B[128 x ldb] + bias[col]
// block = 256 threads = 8 waves; wave -> one 16x64 output tile. M % 16 == 0.
// ---------------------------------------------------------------------------
__global__ __launch_bounds__(256) void rgcn_gemm_root(
    const float* __restrict__ A, const float* __restrict__ B,
    const float* __restrict__ bias, float* __restrict__ C, int M, int ldb) {
  __shared__ float2 Bs[64][BS_STRIDE];
  const int tid = threadIdx.x;
  const int colBase = blockIdx.y * 64;
  load_b_slab(Bs, B, colBase, ldb, tid);
  __syncthreads();

  const int wave = tid >> 5;
  const int lane = tid & 31;
  const int laneLo = lane & 15;
  const int laneHi = lane >> 4;
  const int rowBase = (blockIdx.x * 8 + wave) * 16;
  if (rowBase >= M) return;  // wave-uniform (EXEC all-ones for WMMA)

  v8f acc0 = {}, acc1 = {}, acc2 = {}, acc3 = {};
  const float* Arow = A + (size_t)(rowBase + laneLo) * 128;
  wmma_tile(Arow, Bs, laneLo, laneHi, acc0, acc1, acc2, acc3);

#pragma unroll
  for (int v = 0; v < 8; ++v) {
    const int row = rowBase + v + (laneHi << 3);
    float* Crow = C + (size_t)row * ldb + colBase;
    float vals[4] = {acc0[v], acc1[v], acc2[v], acc3[v]};
#pragma unroll
    for (int j = 0; j < 4; ++j) {
      const int col = laneLo + j * 16;
      Crow[col] = vals[j] + bias[colBase + col];
    }
  }
}

// ---------------------------------------------------------------------------
// Per-relation compacted GEMM:
//   C[list[i]] += invc[list[i]] * (Agg[i] @ W_r)   for i in [0, nnz)
// Agg rows are zero-padded to a 128-row multiple => unguarded A loads.
// Blocks beyond nnz exit immediately (device-side sparsity skip).
// ---------------------------------------------------------------------------
__global__ __launch_bounds__(256) void rgcn_gemm_rel(
    const float* __restrict__ A, const float* __restrict__ B,
    const float* __restrict__ invc, const int* __restrict__ list,
    const int* __restrict__ nnzPtr, float* __restrict__ C, int ldb) {
  const int nnz = *nnzPtr;
  if (blockIdx.x * 128 >= nnz) return;  // block-uniform skip

  __shared__ float2 Bs[64][BS_STRIDE];
  const int tid = threadIdx.x;
  const int colBase = blockIdx.y * 64;
  load_b_slab(Bs, B, colBase, ldb, tid);
  __syncthreads();

  const int wave = tid >> 5;
  const int lane = tid & 31;
  const int laneLo = lane & 15;
  const int laneHi = lane >> 4;
  const int rowBase = blockIdx.x * 128 + wave * 16;
  if (rowBase >= nnz) return;  // wave-uniform; A rows < roundup(nnz,128) are zeroed

  v8f acc0 = {}, acc1 = {}, acc2 = {}, acc3 = {};
  const float* Arow = A + (size_t)(rowBase + laneLo) * 128;
  wmma_tile(Arow, Bs, laneLo, laneHi, acc0, acc1, acc2, acc3);

#pragma unroll
  for (int v = 0; v < 8; ++v) {
    const int row = rowBase + v + (laneHi << 3);
    if (row >= nnz) continue;
    const int node = list[row];
    const float s = invc[node];
    float* Crow = C + (size_t)node * ldb + colBase;
    float vals[4] = {acc0[v], acc1[v], acc2[v], acc3[v]};
#pragma unroll
    for (int j = 0; j < 4; ++j) {
      const int col = laneLo + j * 16;
      Crow[col] += vals[j] * s;
    }
  }
}

// ---------------------------------------------------------------------------
// host launch
// ---------------------------------------------------------------------------
extern "C" void kernel_launch(void* const* d_in, const int* in_sizes, int n_in,
                              void* d_out, int out_size, void* d_ws, size_t ws_size,
                              hipStream_t stream) {
  (void)in_sizes; (void)n_in; (void)out_size; (void)ws_size;

  const float* x      = (const float*)d_in[0];
  const int*   eidx   = (const int*)d_in[1];
  const int*   etype  = (const int*)d_in[2];
  const float* comp1  = (const float*)d_in[3];
  const float* basis1 = (const float*)d_in[4];
  const float* root1  = (const float*)d_in[5];
  const float* bias1  = (const float*)d_in[6];
  const float* comp2  = (const float*)d_in[7];
  const float* basis2 = (const float*)d_in[8];
  const float* root2  = (const float*)d_in[9];
  const float* bias2  = (const float*)d_in[10];
  const int* src = eidx;
  const int* dst = eidx + NN_EDGES;
  float* out = (float*)d_out;

  // workspace carve-up (~154 MB)
  char* ws = (char*)d_ws;
  float* W1 = (float*)ws;   ws += (size_t)N_REL * DIM_IN * DIM_HID * 4;
  float* W2 = (float*)ws;   ws += (size_t)N_REL * DIM_HID * DIM_OUT * 4;
  float* invc = (float*)ws; ws += (size_t)N_REL * NN_NODES * 4;  // int counts, then f32
  int* map = (int*)ws;      ws += (size_t)N_REL * NN_NODES * 4;
  int* list = (int*)ws;     ws += (size_t)N_REL * NN_NODES * 4;
  int* nnz = (int*)ws;      ws += 256;
  float* h = (float*)ws;    ws += (size_t)NN_NODES * DIM_HID * 4;
  float* agg = (float*)ws;  ws += (size_t)NN_NODES * DIM_HID * 4;

  const int TPB = 256;
  // weights from bases
  {
    int n1 = N_REL * DIM_IN * DIM_HID;
    combine_weights<<<(n1 + TPB - 1) / TPB, TPB, 0, stream>>>(comp1, basis1, W1, N_REL, N_BASES, DIM_IN * DIM_HID);
    int n2 = N_REL * DIM_HID * DIM_OUT;
    combine_weights<<<(n2 + TPB - 1) / TPB, TPB, 0, stream>>>(comp2, basis2, W2, N_REL, N_BASES, DIM_HID * DIM_OUT);
  }
  // counts -> compaction (shared by both layers)
  {
    int nc = N_REL * NN_NODES;
    zero_i32<<<(nc + TPB - 1) / TPB, TPB, 0, stream>>>((int*)invc, nc);
    zero_i32<<<1, 64, 0, stream>>>(nnz, N_REL);
    count_edges<<<(NN_EDGES + TPB - 1) / TPB, TPB, 0, stream>>>(dst, etype, (int*)invc, NN_EDGES, NN_NODES);
    build_compact<<<(nc + TPB - 1) / TPB, TPB, 0, stream>>>(invc, map, list, nnz, NN_NODES, nc);
  }

  const int mBlocks = (NN_NODES / 16 + 7) / 8;  // 782 (dense tiles of 128 rows)
  const int scatGrid = (NN_EDGES + 31) / 32;    // 50000 blocks of 1024
  const int aggGrid = (NN_NODES * DIM_HID + TPB - 1) / TPB;

  // ---------------- layer 1: x[100k,128] -> h[100k,128] ----------------
  {
    dim3 g(mBlocks, DIM_HID / 64);
    rgcn_gemm_root<<<g, 256, 0, stream>>>(x, root1, bias1, h, NN_NODES, DIM_HID);
    for (int r = 0; r < N_REL; ++r) {
      zero_agg<<<aggGrid, TPB, 0, stream>>>(agg, nnz + r);
      scatter_rel<<<scatGrid, 1024, 0, stream>>>(src, dst, etype, map, x, agg, NN_EDGES, r);
      rgcn_gemm_rel<<<g, 256, 0, stream>>>(
          agg, W1 + (size_t)r * DIM_IN * DIM_HID, invc + (size_t)r * NN_NODES,
          list + (size_t)r * NN_NODES, nnz + r, h, DIM_HID);
    }
    int nh = NN_NODES * DIM_HID;
    relu_k<<<(nh + TPB - 1) / TPB, TPB, 0, stream>>>(h, nh);
  }

  // ---------------- layer 2: h[100k,128] -> out[100k,64] ----------------
  {
    dim3 g(mBlocks, DIM_OUT / 64);
    rgcn_gemm_root<<<g, 256, 0, stream>>>(h, root2, bias2, out, NN_NODES, DIM_OUT);
    for (int r = 0; r < N_REL; ++r) {
      zero_agg<<<aggGrid, TPB, 0, stream>>>(agg, nnz + r);
      scatter_rel<<<scatGrid, 1024, 0, stream>>>(src, dst, etype, map, h, agg, NN_EDGES, r);
      rgcn_gemm_rel<<<g, 256, 0, stream>>>(
          agg, W2 + (size_t)r * DIM_HID * DIM_OUT, invc + (size_t)r * NN_NODES,
          list + (size_t)r * NN_NODES, nnz + r, out, DIM_OUT);
    }
    int no = NN_NODES * DIM_OUT;
    relu_k<<<(no + TPB - 1) / TPB, TPB, 0, stream>>>(out, no);
  }
}